// HeterogeneousGraphSAGE_36644660970074
// MI455X (gfx1250) — compile-verified
//
#include <hip/hip_runtime.h>
#include <hip/hip_bf16.h>

typedef __attribute__((ext_vector_type(2))) float v2f;
typedef __attribute__((ext_vector_type(8))) float v8f;

#define N_OUT 256

// ---------------------------------------------------------------------------
// Build xg = concat(x_game[64], emb_game[64]) -> [NG, 128]
// ---------------------------------------------------------------------------
__global__ void concat_xg_kernel(const float* __restrict__ xgame,
                                 const float* __restrict__ emb,
                                 float* __restrict__ xg, int total) {
    int i = blockIdx.x * blockDim.x + threadIdx.x;
    if (i >= total) return;
    int node = i >> 7;
    int c = i & 127;
    xg[i] = (c < 64) ? xgame[node * 64 + c] : emb[node * 64 + (c - 64)];
}

// ---------------------------------------------------------------------------
// CSR build: count (+rank), exclusive scan, scatter edge ids
// ---------------------------------------------------------------------------
__global__ void count_edges_kernel(const int* __restrict__ dst, int* __restrict__ cnt,
                                   int* __restrict__ rank, int E) {
    int e = blockIdx.x * blockDim.x + threadIdx.x;
    if (e >= E) return;
    rank[e] = atomicAdd(&cnt[dst[e]], 1);
}

__global__ void scan_exclusive_kernel(int* __restrict__ data, int n) {
    __shared__ int tmp[1024];
    __shared__ int s_carry;
    if (threadIdx.x == 0) s_carry = 0;
    __syncthreads();
    for (int base = 0; base < n; base += 1024) {
        int i = base + (int)threadIdx.x;
        int v = (i < n) ? data[i] : 0;
        tmp[threadIdx.x] = v;
        __syncthreads();
        for (int off = 1; off < 1024; off <<= 1) {
            int t = (threadIdx.x >= (unsigned)off) ? tmp[threadIdx.x - off] : 0;
            __syncthreads();
            tmp[threadIdx.x] += t;
            __syncthreads();
        }
        int carry = s_carry;
        if (i < n) data[i] = carry + tmp[threadIdx.x] - v;   // exclusive
        __syncthreads();
        if (threadIdx.x == 1023) s_carry = carry + tmp[1023];
        __syncthreads();
    }
    if (threadIdx.x == 0) data[n] = s_carry;
}

__global__ void scatter_edges_kernel(const int* __restrict__ dst, const int* __restrict__ off,
                                     const int* __restrict__ rank, int* __restrict__ eid, int E) {
    int e = blockIdx.x * blockDim.x + threadIdx.x;
    if (e >= E) return;
    eid[off[dst[e]] + rank[e]] = e;
}

// ---------------------------------------------------------------------------
// Per-destination weighted segment sum via CSR, float4-vectorized:
//   msg[node][k] = sum_{e in row(node)} xsrc[src[e]][k] * ew[e]
// blockDim.x == K/4 (32 or 64), one block per destination node. Each thread
// owns 4 consecutive features -> global_load_b128 gathers (512B per wave
// VMEM instruction). K=128 case is a single wave: barriers become S_NOP.
// ---------------------------------------------------------------------------
__global__ void conv_gather_kernel(const float4* __restrict__ xsrc4, const float* __restrict__ ew,
                                   const int* __restrict__ srcArr, const int* __restrict__ off,
                                   const int* __restrict__ eid, float4* __restrict__ msg4, int K4) {
    int node = blockIdx.x;
    int beg = off[node], end = off[node + 1];
    int tid = threadIdx.x;
    int chunk = blockDim.x;            // 32 or 64
    __shared__ int s_src[64];
    __shared__ float s_w[64];
    float4 acc = {0.f, 0.f, 0.f, 0.f};
    for (int e0 = beg; e0 < end; e0 += chunk) {
        int cnt = min(chunk, end - e0);
        if (tid == 0 && e0 + chunk < end)
            __builtin_prefetch(&eid[e0 + chunk], 0, 1);   // global_prefetch_b8
        if (tid < cnt) {
            int e = eid[e0 + tid];
            s_src[tid] = srcArr[e];
            s_w[tid] = ew[e];
        }
        __syncthreads();
        for (int j = 0; j < cnt; j++) {
            float4 x = xsrc4[(size_t)s_src[j] * K4 + tid];
            float w = s_w[j];
            acc.x = fmaf(x.x, w, acc.x);
            acc.y = fmaf(x.y, w, acc.y);
            acc.z = fmaf(x.z, w, acc.z);
            acc.w = fmaf(x.w, w, acc.w);
        }
        __syncthreads();
    }
    msg4[(size_t)node * K4 + tid] = acc;
}

// ---------------------------------------------------------------------------
// Fused dual-GEMM epilogue kernel (fp32 WMMA):
//   out[M,256] = A1[M,K1] @ W1[K1,256] + A2[M,K2] @ W2[K2,256] + bias (+ReLU)
// block = 256 threads (8 waves). Each wave owns a 16-row tile x 64 columns
// (4 WMMA n-tiles) so the A fragment is reused 4x per k-step.
// grid = (ceil(M/128), 4)
// ---------------------------------------------------------------------------
__global__ void gemm_dual_kernel(const float* __restrict__ A1, int K1, const float* __restrict__ W1,
                                 const float* __restrict__ A2, int K2, const float* __restrict__ W2,
                                 const float* __restrict__ bias, float* __restrict__ out,
                                 int M, int doRelu) {
    const int wave = threadIdx.x >> 5;
    const int lane = threadIdx.x & 31;
    const int mtile = blockIdx.x * 8 + wave;
    if (mtile * 16 >= M) return;                 // wave-uniform: EXEC stays all-ones
    const int ngrp = blockIdx.y;                 // 64-column group
    const int m = lane & 15;
    const int koff = (lane >> 4) * 2;            // k sub-offset {0,2}
    const int row = mtile * 16 + m;
    const int nbase = ngrp * 64 + m;             // column index for B/C fragments

    v8f acc[4] = {v8f{}, v8f{}, v8f{}, v8f{}};

    // pass 1: message matmul
    const float* arow1 = A1 + (size_t)row * K1;
    for (int kk = 0; kk < K1; kk += 4) {
        v2f a = *(const v2f*)(arow1 + kk + koff);
        const float* w0 = W1 + (size_t)(kk + koff) * N_OUT + nbase;
        const float* w1 = w0 + N_OUT;
#pragma unroll
        for (int t = 0; t < 4; t++) {
            v2f b;
            b.x = w0[t * 16];
            b.y = w1[t * 16];
            acc[t] = __builtin_amdgcn_wmma_f32_16x16x4_f32(
                false, a, false, b, (short)0, acc[t], false, false);
        }
    }
    // pass 2: root matmul
    const float* arow2 = A2 + (size_t)row * K2;
    for (int kk = 0; kk < K2; kk += 4) {
        v2f a = *(const v2f*)(arow2 + kk + koff);
        const float* w0 = W2 + (size_t)(kk + koff) * N_OUT + nbase;
        const float* w1 = w0 + N_OUT;
#pragma unroll
        for (int t = 0; t < 4; t++) {
            v2f b;
            b.x = w0[t * 16];
            b.y = w1[t * 16];
            acc[t] = __builtin_amdgcn_wmma_f32_16x16x4_f32(
                false, a, false, b, (short)0, acc[t], false, false);
        }
    }
    // epilogue: bias (+ relu), store. D layout: VGPR v -> row v + (lane>>4)*8, col lane&15.
    const int rbase = mtile * 16 + ((lane >> 4) * 8);
#pragma unroll
    for (int t = 0; t < 4; t++) {
        int c = ngrp * 64 + t * 16 + m;
        float bv = bias[c];
#pragma unroll
        for (int v = 0; v < 8; v++) {
            float x = acc[t][v] + bv;
            if (doRelu) x = fmaxf(x, 0.f);
            out[(size_t)(rbase + v) * N_OUT + c] = x;
        }
    }
}

// ---------------------------------------------------------------------------
// Decoder: out[s] = dot(u3[row[s]], g3[col[s]]) over 256 (one wave per score)
// ---------------------------------------------------------------------------
__global__ void decode_kernel(const float* __restrict__ u3, const float* __restrict__ g3,
                              const int* __restrict__ rows, const int* __restrict__ cols,
                              float* __restrict__ out, int S) {
    int gwarp = (blockIdx.x * blockDim.x + threadIdx.x) >> 5;
    int lane = threadIdx.x & 31;
    if (gwarp >= S) return;
    const float4* ua = (const float4*)(u3 + (size_t)rows[gwarp] * N_OUT);
    const float4* ga = (const float4*)(g3 + (size_t)cols[gwarp] * N_OUT);
    float acc = 0.f;
#pragma unroll
    for (int it = 0; it < 2; it++) {
        float4 a = ua[lane + it * 32];
        float4 b = ga[lane + it * 32];
        acc += a.x * b.x + a.y * b.y + a.z * b.z + a.w * b.w;
    }
#pragma unroll
    for (int off = 16; off > 0; off >>= 1) acc += __shfl_xor(acc, off, 32);
    if (lane == 0) out[gwarp] = acc;
}

// ---------------------------------------------------------------------------
extern "C" void kernel_launch(void* const* d_in, const int* in_sizes, int n_in,
                              void* d_out, int out_size, void* d_ws, size_t ws_size,
                              hipStream_t stream) {
    const float* xu     = (const float*)d_in[0];
    const float* xgame  = (const float*)d_in[1];
    const float* emb    = (const float*)d_in[2];
    const float* ew_ug  = (const float*)d_in[3];
    const float* ew_gu  = (const float*)d_in[4];
    const int*   src_ug = (const int*)d_in[5];
    const int*   dst_ug = (const int*)d_in[6];
    const int*   src_gu = (const int*)d_in[7];
    const int*   dst_gu = (const int*)d_in[8];
    const int*   srow   = (const int*)d_in[9];
    const int*   scol   = (const int*)d_in[10];

    // per-layer weights: W_rel_ug, b_ug, W_root_g, W_rel_gu, b_gu, W_root_u
    const float *Wrug[3], *bug[3], *Wrtg[3], *Wrgu[3], *bgu[3], *Wrtu[3];
    for (int l = 0; l < 3; l++) {
        int base = 11 + l * 6;
        Wrug[l] = (const float*)d_in[base + 0];
        bug[l]  = (const float*)d_in[base + 1];
        Wrtg[l] = (const float*)d_in[base + 2];
        Wrgu[l] = (const float*)d_in[base + 3];
        bgu[l]  = (const float*)d_in[base + 4];
        Wrtu[l] = (const float*)d_in[base + 5];
    }

    const int NU = in_sizes[0] / 128;   // 100000
    const int NG = in_sizes[1] / 64;    // 20000
    const int E  = in_sizes[3];         // 1000000
    const int S  = in_sizes[9];         // 500000

    // Workspace carve (256B aligned segments)
    char* p = (char*)d_ws;
    auto carve = [&](size_t bytes) -> void* {
        void* r = (void*)p;
        p += (bytes + 255) & ~(size_t)255;
        return r;
    };
    float* XG  = (float*)carve((size_t)NG * 128 * 4);
    float* MSG = (float*)carve((size_t)NU * 256 * 4);
    float* U0  = (float*)carve((size_t)NU * 256 * 4);
    float* U1  = (float*)carve((size_t)NU * 256 * 4);
    float* G0  = (float*)carve((size_t)NG * 256 * 4);
    float* G1  = (float*)carve((size_t)NG * 256 * 4);
    int* OFF_UG = (int*)carve((size_t)(NG + 1) * 4);
    int* OFF_GU = (int*)carve((size_t)(NU + 1) * 4);
    int* EID_UG = (int*)carve((size_t)E * 4);
    int* EID_GU = (int*)carve((size_t)E * 4);
    int* RANK   = (int*)carve((size_t)E * 4);

    const int eblk = (E + 255) / 256;

    // xg = concat(x_game, emb_game)
    {
        int total = NG * 128;
        concat_xg_kernel<<<(total + 255) / 256, 256, 0, stream>>>(xgame, emb, XG, total);
    }

    // CSR for u->g edges (dst = games)
    hipMemsetAsync(OFF_UG, 0, (size_t)(NG + 1) * 4, stream);
    count_edges_kernel<<<eblk, 256, 0, stream>>>(dst_ug, OFF_UG, RANK, E);
    scan_exclusive_kernel<<<1, 1024, 0, stream>>>(OFF_UG, NG);
    scatter_edges_kernel<<<eblk, 256, 0, stream>>>(dst_ug, OFF_UG, RANK, EID_UG, E);
    // CSR for g->u edges (dst = users)
    hipMemsetAsync(OFF_GU, 0, (size_t)(NU + 1) * 4, stream);
    count_edges_kernel<<<eblk, 256, 0, stream>>>(dst_gu, OFF_GU, RANK, E);
    scan_exclusive_kernel<<<1, 1024, 0, stream>>>(OFF_GU, NU);
    scatter_edges_kernel<<<eblk, 256, 0, stream>>>(dst_gu, OFF_GU, RANK, EID_GU, E);

    dim3 gG((NG + 127) / 128, 4);
    dim3 gU((NU + 127) / 128, 4);

    // ---- Layer 1 ----  (K = 128 -> 32 threads/node; K4 = 32)
    conv_gather_kernel<<<NG, 32, 0, stream>>>((const float4*)xu, ew_ug, src_ug, OFF_UG, EID_UG,
                                              (float4*)MSG, 32);
    gemm_dual_kernel<<<gG, 256, 0, stream>>>(MSG, 128, Wrug[0], XG, 128, Wrtg[0], bug[0], G0, NG, 1);
    conv_gather_kernel<<<NU, 32, 0, stream>>>((const float4*)XG, ew_gu, src_gu, OFF_GU, EID_GU,
                                              (float4*)MSG, 32);
    gemm_dual_kernel<<<gU, 256, 0, stream>>>(MSG, 128, Wrgu[0], xu, 128, Wrtu[0], bgu[0], U0, NU, 1);

    // ---- Layer 2 ----  (K = 256 -> 64 threads/node; K4 = 64)
    conv_gather_kernel<<<NG, 64, 0, stream>>>((const float4*)U0, ew_ug, src_ug, OFF_UG, EID_UG,
                                              (float4*)MSG, 64);
    gemm_dual_kernel<<<gG, 256, 0, stream>>>(MSG, 256, Wrug[1], G0, 256, Wrtg[1], bug[1], G1, NG, 1);
    conv_gather_kernel<<<NU, 64, 0, stream>>>((const float4*)G0, ew_gu, src_gu, OFF_GU, EID_GU,
                                              (float4*)MSG, 64);
    gemm_dual_kernel<<<gU, 256, 0, stream>>>(MSG, 256, Wrgu[1], U0, 256, Wrtu[1], bgu[1], U1, NU, 1);

    // ---- Layer 3 (no ReLU) ----  g3 -> G0, u3 -> U0 (ping-pong reuse)
    conv_gather_kernel<<<NG, 64, 0, stream>>>((const float4*)U1, ew_ug, src_ug, OFF_UG, EID_UG,
                                              (float4*)MSG, 64);
    gemm_dual_kernel<<<gG, 256, 0, stream>>>(MSG, 256, Wrug[2], G1, 256, Wrtg[2], bug[2], G0, NG, 0);
    conv_gather_kernel<<<NU, 64, 0, stream>>>((const float4*)G1, ew_gu, src_gu, OFF_GU, EID_GU,
                                              (float4*)MSG, 64);
    gemm_dual_kernel<<<gU, 256, 0, stream>>>(MSG, 256, Wrgu[2], U1, 256, Wrtu[2], bgu[2], U0, NU, 0);

    // ---- Decoder ----
    decode_kernel<<<(S + 7) / 8, 256, 0, stream>>>(U0, G0, srow, scol, (float*)d_out, S);
}